// SPN_18906446037566
// MI455X (gfx1250) — compile-verified
//
#include <hip/hip_runtime.h>
#include <math.h>

// ---------------------------------------------------------------------------
// Progressive width-1 MLP chain (B=65536, IN_F=256, 128 layers, out=last 32).
//
// Strategy (MI455X / gfx1250, wave32):
//  * Layers blocked in groups of 16. Inter-group contribution is a dense
//    GEMM done with V_WMMA_F32_16X16X4_F32 (fp32 in/out -> matches the fp32
//    reference; cascading exact-GELU chain is too precision-sensitive for
//    bf16/f16 inputs).
//  * In-group 16x16 lower triangle (sequential, GELU interleaved) done
//    row-per-lane in registers (fully unrolled, 120 FMAs + 16 erff).
//  * One wave (32 threads) per 32 batch rows; the growing [32 x 384] fp32
//    activation buffer lives in LDS in a column-pair-major, padded layout
//    (stride 66 floats per column pair -> bank-conflict-free ds_load_b64
//    WMMA A-fragments and conflict-free x staging).
//  * Single-wave workgroup: __syncthreads() degenerates to the required
//    s_wait_dscnt (barrier ops are S_NOP for 1-wave groups per ISA), giving
//    cheap cross-lane LDS ordering.
// ---------------------------------------------------------------------------

typedef float v8f __attribute__((ext_vector_type(8)));
typedef float v4f __attribute__((ext_vector_type(4)));
typedef float v2f __attribute__((ext_vector_type(2)));

#define IN_F     256
#define TOTAL_L  128
#define OUT_N    32
#define HIDDEN_L 96
#define DW       384                  // IN_F + TOTAL_L
#define ROWS     32                   // batch rows per wave/block
#define NCP      (DW / 2)             // 192 column pairs
#define CPSTR    66                   // floats per column pair (32*2 + 2 pad)

__device__ __forceinline__ float gelu_exact(float v) {
    return 0.5f * v * (1.0f + erff(v * 0.70710678118654752440f));
}

__global__ __launch_bounds__(32) void spn_chain_kernel(
    const float* __restrict__ x,      // [B, 256]
    const float* __restrict__ W,      // [128, 384]
    const float* __restrict__ bias,   // [128]
    float* __restrict__ out)          // [B, 32]
{
    __shared__ float buf[NCP * CPSTR];   // 50688 B: [colpair][row] as float2
    __shared__ float sAcc[16 * 33];      //  2112 B: staged 16x32 acc tile (pad 33)

    const int lane    = threadIdx.x;          // 0..31
    const int rowBase = blockIdx.x * ROWS;
    const int half    = lane >> 4;            // K-half select for WMMA frags
    const int nIdx    = lane & 15;            // M (A) / N (B) index

    // ---- stage x tile: 32 rows x 256 cols -> column-pair-major LDS --------
    for (int r = 0; r < ROWS; ++r) {
        const float* xr = x + (size_t)(rowBase + r) * IN_F;
        #pragma unroll
        for (int q = 0; q < 4; ++q) {
            const int cp = lane + 32 * q;                    // colpair 0..127
            v2f v = *(const v2f*)(xr + 2 * cp);              // coalesced b64
            *(v2f*)(&buf[cp * CPSTR + 2 * r]) = v;           // conflict-free
        }
    }
    __syncthreads();

    for (int g = 0; g < 8; ++g) {
        const int i0 = 16 * g;
        const int K  = IN_F + i0;           // dense K for this group

        // ---- dense part: acc[32x16] = buf[:, :K] * W[i0:i0+16, :K]^T ------
        v8f acc0 = {};                      // rows 0..15
        v8f acc1 = {};                      // rows 16..31
        const float* wB = W + (size_t)(i0 + nIdx) * DW + 2 * half;

        for (int k = 0; k < K; k += 4) {
            // A frag: lane -> row = nIdx (+16 for tile1), cols k+2*half..+1
            const int cp = (k >> 1) + half;
            v2f a0 = *(const v2f*)(&buf[cp * CPSTR + 2 * nIdx]);
            v2f a1 = *(const v2f*)(&buf[cp * CPSTR + 2 * (nIdx + 16)]);
            // B frag: lane -> N = nIdx, K rows k+2*half..+1 (contig in W row)
            v2f bf = *(const v2f*)(wB + k);
            acc0 = __builtin_amdgcn_wmma_f32_16x16x4_f32(
                false, a0, false, bf, (short)0, acc0, false, false);
            acc1 = __builtin_amdgcn_wmma_f32_16x16x4_f32(
                false, a1, false, bf, (short)0, acc1, false, false);
        }

        // bias: lane holds column n = nIdx across all 8 C VGPRs
        const float bv = bias[i0 + nIdx];
        #pragma unroll
        for (int r = 0; r < 8; ++r) { acc0[r] += bv; acc1[r] += bv; }

        // ---- stage acc to LDS: sAcc[n][row], C layout: M = r + 8*half -----
        #pragma unroll
        for (int r = 0; r < 8; ++r) {
            sAcc[nIdx * 33 + (8 * half + r)]      = acc0[r];
            sAcc[nIdx * 33 + 16 + (8 * half + r)] = acc1[r];
        }
        __syncthreads();

        // ---- in-group triangle, row-per-lane ------------------------------
        float p[16];
        #pragma unroll
        for (int n = 0; n < 16; ++n) p[n] = sAcc[n * 33 + lane];

        float a[16];
        float og[16];
        #pragma unroll
        for (int n = 0; n < 16; ++n) {
            float v = p[n];
            const float* wr = W + (size_t)(i0 + n) * DW + IN_F + i0; // uniform
            #pragma unroll
            for (int m = 0; m < n; ++m) v += a[m] * wr[m];           // s_load w
            const int layer = i0 + n;
            const float actv = (layer < HIDDEN_L) ? gelu_exact(v) : v;
            a[n]  = actv;
            og[n] = actv;                     // layers >= 96: actv == v
            const int col = IN_F + i0 + n;    // append to growing buffer
            buf[(col >> 1) * CPSTR + 2 * lane + (col & 1)] = actv;
        }
        __syncthreads();

        // ---- emit output for layers 96..127 (groups 6,7) ------------------
        if (g >= 6) {
            float* op = out + (size_t)(rowBase + lane) * OUT_N + (g - 6) * 16;
            #pragma unroll
            for (int n = 0; n < 16; n += 4) {
                v4f t = { og[n], og[n + 1], og[n + 2], og[n + 3] };
                *(v4f*)(op + n) = t;          // 16B-aligned, contig per row
            }
        }
    }
}

extern "C" void kernel_launch(void* const* d_in, const int* in_sizes, int n_in,
                              void* d_out, int out_size, void* d_ws, size_t ws_size,
                              hipStream_t stream) {
    const float* x    = (const float*)d_in[0];   // [B, 256]
    const float* W    = (const float*)d_in[1];   // [128, 384]
    const float* bias = (const float*)d_in[2];   // [128]
    float* out        = (float*)d_out;           // [B, 32]

    const int B = in_sizes[0] / IN_F;            // 65536
    const int grid = B / ROWS;                   // 2048 blocks of 1 wave
    spn_chain_kernel<<<grid, ROWS, 0, stream>>>(x, W, bias, out);
}